// AIGer_50775103373990
// MI455X (gfx1250) — compile-verified
//
#include <hip/hip_runtime.h>
#include <math.h>

typedef float v2f __attribute__((ext_vector_type(2)));
typedef float v8f __attribute__((ext_vector_type(8)));

// ---------------------------------------------------------------------------
// Dense GEMM  C[nrows x NCOLS] = act(A[nrows x K] @ W[K x NCOLS] + bias)
// using V_WMMA_F32_16X16X4_F32 (f32 A/B/C, 16x16 tile, K-step 4).
// - One wave per 16x16 output tile; nrows % 16 == 0 (100000 is).
// - W staged in LDS once per block; K-loop fully unrolled via templates so
//   all loads use immediate offsets (no per-iteration address math).
// ACT: 0 = identity, 1 = tanh
// ---------------------------------------------------------------------------
template <int K, int NCOLS, int ACT>
__global__ void gemm_wmma_f32(const float* __restrict__ A,
                              const float* __restrict__ W,
                              const float* __restrict__ bias,
                              float* __restrict__ C, int nrows)
{
    __shared__ float sW[K * NCOLS];
#pragma unroll
    for (int i = threadIdx.x; i < K * NCOLS; i += 128)
        sW[i] = W[i];
    __syncthreads();

    const int lane = threadIdx.x & 31;
    const int wave = blockIdx.x * (blockDim.x >> 5) + (threadIdx.x >> 5);
    constexpr int colTiles = NCOLS / 16;
    const int rowTile = wave / colTiles;
    const int colTile = wave - rowTile * colTiles;
    const int row0 = rowTile << 4;
    if (row0 >= nrows) return;                 // wave-uniform guard
    const int col0 = colTile << 4;
    const int lr = lane & 15;                  // row (A) / col (B,C) in tile
    const int lh = lane >> 4;                  // K-half selector

    v8f c = {0.f, 0.f, 0.f, 0.f, 0.f, 0.f, 0.f, 0.f};

    // A layout (16x4 f32): v0 holds K=(0|2), v1 holds K=(1|3) per lane-half
    const float* Arow = A + (size_t)(row0 + lr) * K + 2 * lh;
    // B layout (4x16 f32): v0 holds W row K=(0|2) at column col0+lr
    const float* sWp = sW + 2 * lh * NCOLS + col0 + lr;

#pragma unroll
    for (int k0 = 0; k0 < K; k0 += 4) {
        v2f a, b;
        a.x = Arow[k0];                        // b64 load, immediate offset
        a.y = Arow[k0 + 1];
        b.x = sWp[k0 * NCOLS];                 // ds_load, immediate offset
        b.y = sWp[(k0 + 1) * NCOLS];
        c = __builtin_amdgcn_wmma_f32_16x16x4_f32(false, a, false, b,
                                                  (short)0, c, false, false);
    }

    const float bb = bias[col0 + lr];
    float* Cp = C + (size_t)(row0 + 8 * lh) * NCOLS + col0 + lr;
#pragma unroll
    for (int r = 0; r < 8; ++r) {
        float v = c[r] + bb;
        if (ACT) v = tanhf(v);
        Cp[r * NCOLS] = v;
    }
}

// ---------------------------------------------------------------------------
// Edge scatter: agg[dst] += T[src]  (32 floats/edge, float4-vectorized loads,
// f32 global atomics resolve in L2 — agg is only 12.8 MB, L2 is 192 MB)
// ---------------------------------------------------------------------------
__global__ void scatter_add32(const int* __restrict__ src,
                              const int* __restrict__ dst,
                              const float* __restrict__ T,
                              float* __restrict__ agg, int E)
{
    int t = blockIdx.x * blockDim.x + threadIdx.x;
    int e = t >> 3;
    if (e >= E) return;
    int j = (t & 7) << 2;
    int s = src[e];
    int d = dst[e];
    float4 v = *reinterpret_cast<const float4*>(T + (size_t)s * 32 + j);
    float* p = agg + (size_t)d * 32 + j;
    atomicAdd(p + 0, v.x);
    atomicAdd(p + 1, v.y);
    atomicAdd(p + 2, v.z);
    atomicAdd(p + 3, v.w);
}

__global__ void tanh_inplace(float* __restrict__ x, int n)
{
    int i = blockIdx.x * blockDim.x + threadIdx.x;
    if (i < n) x[i] = tanhf(x[i]);
}

// ---------------------------------------------------------------------------
// Column sum / sum-of-squares for BatchNorm over 64 columns.
// stats[0..63] = sum, stats[64..127] = sumsq (zeroed beforehand).
// ---------------------------------------------------------------------------
__global__ void colstats64(const float* __restrict__ h,
                           float* __restrict__ stats, int nrows)
{
    __shared__ float ssum[256];
    __shared__ float ssq[256];
    const int col = threadIdx.x & 63;
    const int rt  = threadIdx.x >> 6;          // 0..3
    float s = 0.f, q = 0.f;
    for (int r = blockIdx.x * 4 + rt; r < nrows; r += gridDim.x * 4) {
        float v = h[(size_t)r * 64 + col];
        s += v;
        q += v * v;
    }
    ssum[threadIdx.x] = s;
    ssq[threadIdx.x]  = q;
    __syncthreads();
    if (threadIdx.x < 64) {
        s = ssum[threadIdx.x] + ssum[threadIdx.x + 64] +
            ssum[threadIdx.x + 128] + ssum[threadIdx.x + 192];
        q = ssq[threadIdx.x] + ssq[threadIdx.x + 64] +
            ssq[threadIdx.x + 128] + ssq[threadIdx.x + 192];
        atomicAdd(&stats[col], s);
        atomicAdd(&stats[64 + col], q);
    }
}

__global__ void bn_relu64(const float* __restrict__ h,
                          const float* __restrict__ stats,
                          const float* __restrict__ g,
                          const float* __restrict__ b,
                          float* __restrict__ out, int n, float invN)
{
    int i = blockIdx.x * blockDim.x + threadIdx.x;
    if (i >= n) return;
    int col = i & 63;
    float m = stats[col] * invN;
    float v = stats[64 + col] * invN - m * m;
    float y = (h[i] - m) * rsqrtf(v + 1e-5f) * g[col] + b[col];
    out[i] = y > 0.f ? y : 0.f;
}

// prob = sigmoid(h[N x 64] @ W3[64 x 1] + b3)
__global__ void final_row(const float* __restrict__ h,
                          const float* __restrict__ W3,
                          const float* __restrict__ b3,
                          float* __restrict__ prob, int nrows)
{
    int r = blockIdx.x * blockDim.x + threadIdx.x;
    if (r >= nrows) return;
    const float* row = h + (size_t)r * 64;
    float acc = b3[0];
#pragma unroll
    for (int i = 0; i < 64; ++i) acc += row[i] * W3[i];
    prob[r] = 1.f / (1.f + expf(-acc));
}

// ---------------------------------------------------------------------------
extern "C" void kernel_launch(void* const* d_in, const int* in_sizes, int n_in,
                              void* d_out, int out_size, void* d_ws, size_t ws_size,
                              hipStream_t stream)
{
    const float* init_emb = (const float*)d_in[0];
    const int*   pos      = (const int*)d_in[1];
    const int*   neg      = (const int*)d_in[2];
    const float* c1_Wp = (const float*)d_in[3];  const float* c1_bp = (const float*)d_in[4];
    const float* c1_Wn = (const float*)d_in[5];  const float* c1_bn = (const float*)d_in[6];
    const float* c1_Ws = (const float*)d_in[7];  const float* c1_bs = (const float*)d_in[8];
    const float* c2_Wp = (const float*)d_in[9];  const float* c2_bp = (const float*)d_in[10];
    const float* c2_Wn = (const float*)d_in[11]; const float* c2_bn = (const float*)d_in[12];
    const float* c2_Ws = (const float*)d_in[13]; const float* c2_bs = (const float*)d_in[14];
    const float* w_W  = (const float*)d_in[15];  const float* w_b  = (const float*)d_in[16];
    const float* m1_W = (const float*)d_in[17];  const float* m1_b = (const float*)d_in[18];
    const float* g1   = (const float*)d_in[19];  const float* b1   = (const float*)d_in[20];
    const float* m2_W = (const float*)d_in[21];  const float* m2_b = (const float*)d_in[22];
    const float* g2   = (const float*)d_in[23];  const float* b2   = (const float*)d_in[24];
    const float* m3_W = (const float*)d_in[25];  const float* m3_b = (const float*)d_in[26];

    const int N = in_sizes[0] / 64;     // 100000
    const int E = in_sizes[1] / 2;      // 1000000
    const int* pos_src = pos;       const int* pos_dst = pos + E;
    const int* neg_src = neg;       const int* neg_dst = neg + E;

    // workspace layout (floats)
    float* ws    = (float*)d_ws;
    float* Tp    = ws;                       // N*32
    float* Tn    = Tp + (size_t)N * 32;      // N*32
    float* agg1  = Tn + (size_t)N * 32;      // N*32  -> z1 (tanh in place)
    float* agg2  = agg1 + (size_t)N * 32;    // N*32  -> z2
    float* hA    = agg2 + (size_t)N * 32;    // N*64
    float* hB    = hA + (size_t)N * 64;      // N*64
    float* stats = hB + (size_t)N * 64;      // 128 floats

    float* z_out    = (float*)d_out;             // N*64
    float* prob_out = (float*)d_out + (size_t)N * 64;

    const int rowTiles = N / 16;                 // 6250
    const int grid32 = (rowTiles * 2 + 3) / 4;   // NCOLS=32 -> 2 col tiles
    const int grid64 = (rowTiles * 4 + 3) / 4;   // NCOLS=64 -> 4 col tiles

    // ---- conv1: agg1 = x@Ws+bs ; Tp = x@Wp+bp ; Tn = x@Wn+bn ----------------
    gemm_wmma_f32<64, 32, 0><<<grid32, 128, 0, stream>>>(init_emb, c1_Ws, c1_bs, agg1, N);
    gemm_wmma_f32<64, 32, 0><<<grid32, 128, 0, stream>>>(init_emb, c1_Wp, c1_bp, Tp,   N);
    gemm_wmma_f32<64, 32, 0><<<grid32, 128, 0, stream>>>(init_emb, c1_Wn, c1_bn, Tn,   N);
    {
        int threads = E * 8, blk = 256, grd = (threads + blk - 1) / blk;
        scatter_add32<<<grd, blk, 0, stream>>>(pos_src, pos_dst, Tp, agg1, E);
        scatter_add32<<<grd, blk, 0, stream>>>(neg_src, neg_dst, Tn, agg1, E);
    }
    tanh_inplace<<<(N * 32 + 255) / 256, 256, 0, stream>>>(agg1, N * 32);   // z1

    // ---- conv2 ---------------------------------------------------------------
    gemm_wmma_f32<32, 32, 0><<<grid32, 128, 0, stream>>>(agg1, c2_Ws, c2_bs, agg2, N);
    gemm_wmma_f32<32, 32, 0><<<grid32, 128, 0, stream>>>(agg1, c2_Wp, c2_bp, Tp,   N);
    gemm_wmma_f32<32, 32, 0><<<grid32, 128, 0, stream>>>(agg1, c2_Wn, c2_bn, Tn,   N);
    {
        int threads = E * 8, blk = 256, grd = (threads + blk - 1) / blk;
        scatter_add32<<<grd, blk, 0, stream>>>(pos_src, pos_dst, Tp, agg2, E);
        scatter_add32<<<grd, blk, 0, stream>>>(neg_src, neg_dst, Tn, agg2, E);
    }
    tanh_inplace<<<(N * 32 + 255) / 256, 256, 0, stream>>>(agg2, N * 32);   // z2

    // ---- z = tanh(z2 @ w_W + w_b)  -> first output ---------------------------
    gemm_wmma_f32<32, 64, 1><<<grid64, 128, 0, stream>>>(agg2, w_W, w_b, z_out, N);

    // ---- readout MLP ---------------------------------------------------------
    const float invN = 1.f / (float)N;

    gemm_wmma_f32<64, 64, 0><<<grid64, 128, 0, stream>>>(z_out, m1_W, m1_b, hA, N);
    hipMemsetAsync(stats, 0, 128 * sizeof(float), stream);
    colstats64<<<512, 256, 0, stream>>>(hA, stats, N);
    bn_relu64<<<(N * 64 + 255) / 256, 256, 0, stream>>>(hA, stats, g1, b1, hB, N * 64, invN);

    gemm_wmma_f32<64, 64, 0><<<grid64, 128, 0, stream>>>(hB, m2_W, m2_b, hA, N);
    hipMemsetAsync(stats, 0, 128 * sizeof(float), stream);
    colstats64<<<512, 256, 0, stream>>>(hA, stats, N);
    bn_relu64<<<(N * 64 + 255) / 256, 256, 0, stream>>>(hA, stats, g2, b2, hB, N * 64, invN);

    final_row<<<(N + 255) / 256, 256, 0, stream>>>(hB, m3_W, m3_b, prob_out, N);
}